// DualBranchGNN_Deep_22411139351102
// MI455X (gfx1250) — compile-verified
//
#include <hip/hip_runtime.h>
#include <math.h>

// ---------------------------------------------------------------------------
// DualBranchGNN for MI455X (gfx1250). fp32 throughout; dense GEMMs use
// V_WMMA_F32_16X16X4_F32 (one 16x16 tile per wave32), with all shapes
// compile-time templated so the K-loop fully unrolls into back-to-back WMMA
// with immediate-offset global loads (no per-iteration 64-bit address math).
// Irregular segment ops (GAT softmax, scatter-adds, GCN norm) use f32 atomics
// that stay L2-resident (node tables are 12.8MB << 192MB L2).
// GAT bias is skipped: it cancels exactly in BatchNorm.
// ---------------------------------------------------------------------------

#define Nn   50000
#define Ee   800000
#define ESL  (Ee + Nn)        // edges + self-loops
#define SIG2 900.0f           // SIGMA^2
#define EPSV 1e-5f

typedef __attribute__((ext_vector_type(2))) float v2f;
typedef __attribute__((ext_vector_type(8))) float v8f;

__device__ __forceinline__ float lrelu02(float x){ return x > 0.f ? x : 0.2f * x; }
__device__ __forceinline__ float eluf(float x){ return x > 0.f ? x : (expf(x) - 1.f); }

// float atomic max via monotonic int mapping (sign-dependent op)
__device__ __forceinline__ void atomicMaxF(float* a, float v){
  if (v >= 0.f) atomicMax((int*)a, __float_as_int(v));
  else          atomicMin((unsigned int*)a, __float_as_uint(v));
}

// -------------------------------- utility ----------------------------------
__global__ void k_fill(float* __restrict__ p, float v, size_t n){
  size_t i = (size_t)blockIdx.x * blockDim.x + threadIdx.x;
  if (i < n) p[i] = v;
}

// ------------------------- edge weight layer -------------------------------
__global__ void k_edgew(const float* __restrict__ ef, const float* __restrict__ w1,
                        const float* __restrict__ b1, const float* __restrict__ w2,
                        const float* __restrict__ b2, float* __restrict__ ew){
  int e = blockIdx.x * blockDim.x + threadIdx.x;
  if (e >= Ee) return;
  float f[10];
#pragma unroll
  for (int i = 0; i < 10; ++i) f[i] = ef[(size_t)e * 10 + i];
  float ind = b2[0];
#pragma unroll
  for (int j = 0; j < 10; ++j){
    float h = b1[j];
#pragma unroll
    for (int i = 0; i < 10; ++i) h = fmaf(f[i], w1[i * 10 + j], h);
    ind = fmaf(eluf(h), w2[j], ind);
  }
  ew[e] = expf(-(ind * ind) / SIG2);
}

// ----------------------- WMMA fp32 GEMM (16x16 tile/wave) ------------------
// C[Nn,KO] (= or +=) act(A[Nn,K] @ W[K,KO] + bias)
// Compile-time K/KO/MODE/ACC: K-loop fully unrolls -> K/4 back-to-back
// v_wmma_f32_16x16x4_f32 with immediate-offset loads.
// MODE: 0=none 1=relu 2=sigmoid ; ACC: C += result
template<int K, int KO, int MODE, int ACC>
__global__ void k_gemm(const float* __restrict__ A,
                       const float* __restrict__ W,
                       const float* __restrict__ bias,
                       float* __restrict__ C){
  constexpr int TILES_M = Nn / 16;
  constexpr int TILES_N = KO / 16;
  int wv   = (int)((blockIdx.x * blockDim.x + threadIdx.x) >> 5);
  int lane = threadIdx.x & 31;
  if (wv >= TILES_M * TILES_N) return;      // wave-uniform exit (EXEC stays all-1s)
  int tm = wv / TILES_N, tn = wv - tm * TILES_N;
  int half = lane >> 4, l16 = lane & 15;

  // A frag: lane(l16,half) holds A[tm*16+l16][k+2h], A[..][k+2h+1] (v2f load)
  const float* Ar = A + (size_t)(tm * 16 + l16) * K + 2 * half;
  // B frag: lane(l16,half) holds W[k+2h][col], W[k+2h+1][col]
  const float* Wc = W + (size_t)(2 * half) * KO + tn * 16 + l16;
  __builtin_prefetch(Wc, 0, 1);             // tiny all-wave-shared weights -> GL2

  v8f c = {0.f, 0.f, 0.f, 0.f, 0.f, 0.f, 0.f, 0.f};
#pragma unroll
  for (int k = 0; k < K; k += 4){
    v2f a = *(const v2f*)(Ar + k);
    v2f b;
    b.x = Wc[k * KO];
    b.y = Wc[(k + 1) * KO];
    c = __builtin_amdgcn_wmma_f32_16x16x4_f32(false, a, false, b, (short)0, c,
                                              false, false);
  }

  int col  = tn * 16 + l16;
  int row0 = tm * 16 + half * 8;            // C layout: VGPR r -> M=row0+r, N=col
  float bv = bias ? bias[col] : 0.f;
#pragma unroll
  for (int r = 0; r < 8; ++r){
    float v = c[r] + bv;
    if      (MODE == 1) v = fmaxf(v, 0.f);
    else if (MODE == 2) v = 1.f / (1.f + expf(-v));
    size_t idx = (size_t)(row0 + r) * KO + col;
    if (ACC) v += C[idx];
    C[idx] = v;
  }
}

// --------------------------- GAT attention ---------------------------------
__global__ void k_scores(const float* __restrict__ z, const float* __restrict__ asrc,
                         const float* __restrict__ adst, float* __restrict__ s,
                         float* __restrict__ t, float* __restrict__ mx){
  int n = blockIdx.x * blockDim.x + threadIdx.x;
  if (n >= Nn) return;
  const float* zr = z + (size_t)n * 64;
  float ss = 0.f, tt = 0.f;
#pragma unroll 8
  for (int j = 0; j < 64; ++j){ float v = zr[j]; ss = fmaf(v, asrc[j], ss); tt = fmaf(v, adst[j], tt); }
  s[n] = ss; t[n] = tt;
  mx[n] = lrelu02(ss + tt);                 // init running max with self-loop score
}

__global__ void k_attnmax(const int* __restrict__ src, const int* __restrict__ dst,
                          const float* __restrict__ s, const float* __restrict__ t,
                          float* __restrict__ mx){
  int e = blockIdx.x * blockDim.x + threadIdx.x;
  if (e >= Ee) return;
  atomicMaxF(mx + dst[e], lrelu02(s[src[e]] + t[dst[e]]));
}

__global__ void k_attnexp(const int* __restrict__ src, const int* __restrict__ dst,
                          const float* __restrict__ s, const float* __restrict__ t,
                          const float* __restrict__ mx, float* __restrict__ exb,
                          float* __restrict__ dn){
  int e = blockIdx.x * blockDim.x + threadIdx.x;
  if (e >= ESL) return;
  int si, di;
  if (e < Ee){ si = src[e]; di = dst[e]; } else { si = di = e - Ee; }
  float ex = expf(lrelu02(s[si] + t[di]) - mx[di]);
  exb[e] = ex;
  atomicAdd(dn + di, ex);
}

__global__ void k_gatscatter(const int* __restrict__ src, const int* __restrict__ dst,
                             const float* __restrict__ exb, const float* __restrict__ dn,
                             const float* __restrict__ z, float* __restrict__ agg){
  size_t i = (size_t)blockIdx.x * blockDim.x + threadIdx.x;
  if (i >= (size_t)ESL * 64) return;
  int e = (int)(i >> 6), f = (int)(i & 63);
  int si, di;
  if (e < Ee){ si = src[e]; di = dst[e]; } else { si = di = e - Ee; }
  float a = exb[e] / (dn[di] + 1e-16f);
  atomicAdd(agg + (size_t)di * 64 + f, a * z[(size_t)si * 64 + f]);
}

// ------------------------------ BatchNorm ----------------------------------
__global__ void k_colstats(const float* __restrict__ x, float* __restrict__ cs,
                           float* __restrict__ cq){
  __shared__ float ls[64], lq[64];
  if (threadIdx.x < 64){ ls[threadIdx.x] = 0.f; lq[threadIdx.x] = 0.f; }
  __syncthreads();
  size_t total = (size_t)Nn * 64;
  for (size_t i = (size_t)blockIdx.x * blockDim.x + threadIdx.x; i < total;
       i += (size_t)gridDim.x * blockDim.x){
    float v = x[i]; int c = (int)(i & 63);
    atomicAdd(&ls[c], v); atomicAdd(&lq[c], v * v);
  }
  __syncthreads();
  if (threadIdx.x < 64){
    atomicAdd(cs + threadIdx.x, ls[threadIdx.x]);
    atomicAdd(cq + threadIdx.x, lq[threadIdx.x]);
  }
}

__global__ void k_bnfin(const float* __restrict__ cs, const float* __restrict__ cq,
                        const float* __restrict__ g, const float* __restrict__ b,
                        float* __restrict__ scale, float* __restrict__ shift){
  int c = threadIdx.x;
  if (c >= 64) return;
  float mu  = cs[c] * (1.f / (float)Nn);
  float var = cq[c] * (1.f / (float)Nn) - mu * mu;
  float sc  = g[c] * rsqrtf(var + EPSV);
  scale[c] = sc; shift[c] = b[c] - mu * sc;
}

__global__ void k_bnelu(const float* __restrict__ x, const float* __restrict__ scale,
                        const float* __restrict__ shift, float* __restrict__ y){
  size_t i = (size_t)blockIdx.x * blockDim.x + threadIdx.x;
  if (i >= (size_t)Nn * 64) return;
  int c = (int)(i & 63);
  y[i] = eluf(fmaf(x[i], scale[c], shift[c]));
}

// ----------------------------- GCN branch ----------------------------------
__global__ void k_deg(const int* __restrict__ dst, const float* __restrict__ ew,
                      float* __restrict__ deg){
  int e = blockIdx.x * blockDim.x + threadIdx.x;
  if (e >= Ee) return;
  atomicAdd(deg + dst[e], ew[e]);
}

__global__ void k_dinv(const float* __restrict__ deg, float* __restrict__ dinv){
  int n = blockIdx.x * blockDim.x + threadIdx.x;
  if (n >= Nn) return;
  float d = deg[n];
  dinv[n] = d > 0.f ? rsqrtf(d) : 0.f;
}

__global__ void k_gcnscatter(const int* __restrict__ src, const int* __restrict__ dst,
                             const float* __restrict__ ew, const float* __restrict__ dinv,
                             const float* __restrict__ zg, float* __restrict__ hg){
  size_t i = (size_t)blockIdx.x * blockDim.x + threadIdx.x;
  if (i >= (size_t)ESL * 64) return;
  int e = (int)(i >> 6), f = (int)(i & 63);
  int si, di; float w;
  if (e < Ee){ si = src[e]; di = dst[e]; w = ew[e]; } else { si = di = e - Ee; w = 1.f; }
  float coef = w * dinv[si] * dinv[di];
  atomicAdd(hg + (size_t)di * 64 + f, coef * zg[(size_t)si * 64 + f]);
}

// LayerNorm(+gcn bias fold) + ReLU, one wave32 per node, 2 features/lane
__global__ void k_lnrelu(float* __restrict__ hg, const float* __restrict__ gb,
                         const float* __restrict__ g, const float* __restrict__ b){
  int wv   = (int)((blockIdx.x * blockDim.x + threadIdx.x) >> 5);
  int lane = threadIdx.x & 31;
  if (wv >= Nn) return;
  float* r = hg + (size_t)wv * 64;
  float v0 = r[lane]      + gb[lane];
  float v1 = r[lane + 32] + gb[lane + 32];
  float s = v0 + v1;
#pragma unroll
  for (int o = 16; o > 0; o >>= 1) s += __shfl_xor(s, o, 32);
  float mu = s * (1.f / 64.f);
  float d0 = v0 - mu, d1 = v1 - mu;
  float q = d0 * d0 + d1 * d1;
#pragma unroll
  for (int o = 16; o > 0; o >>= 1) q += __shfl_xor(q, o, 32);
  float rs = rsqrtf(q * (1.f / 64.f) + EPSV);
  r[lane]      = fmaxf(fmaf(g[lane]      * d0, rs, b[lane]),      0.f);
  r[lane + 32] = fmaxf(fmaf(g[lane + 32] * d1, rs, b[lane + 32]), 0.f);
}

// ------------------------------- fusion ------------------------------------
__global__ void k_cat(const float* __restrict__ hl, const float* __restrict__ hg,
                      float* __restrict__ cat){
  size_t i = (size_t)blockIdx.x * blockDim.x + threadIdx.x;
  if (i >= (size_t)Nn * 128) return;
  int n = (int)(i >> 7), c = (int)(i & 127);
  cat[i] = (c < 64) ? hl[(size_t)n * 64 + c] : hg[(size_t)n * 64 + c - 64];
}

__global__ void k_fuse(const float* __restrict__ gate, const float* __restrict__ hl,
                       const float* __restrict__ hg, float* __restrict__ fused){
  size_t i = (size_t)blockIdx.x * blockDim.x + threadIdx.x;
  if (i >= (size_t)Nn * 64) return;
  float gt = gate[i];
  fused[i] = gt * hl[i] + (1.f - gt) * hg[i];
}

__global__ void k_out(const float* __restrict__ fused, const float* __restrict__ w,
                      const float* __restrict__ b, float* __restrict__ pred){
  int wv   = (int)((blockIdx.x * blockDim.x + threadIdx.x) >> 5);
  int lane = threadIdx.x & 31;
  if (wv >= Nn) return;
  const float* r = fused + (size_t)wv * 64;
  float s = fmaf(r[lane], w[lane], r[lane + 32] * w[lane + 32]);
#pragma unroll
  for (int o = 16; o > 0; o >>= 1) s += __shfl_xor(s, o, 32);
  if (lane == 0) pred[wv] = s + b[0];
}

// ---------------------------------------------------------------------------
extern "C" void kernel_launch(void* const* d_in, const int* in_sizes, int n_in,
                              void* d_out, int out_size, void* d_ws, size_t ws_size,
                              hipStream_t stream) {
  (void)in_sizes; (void)n_in; (void)out_size; (void)ws_size;
  const float* x_local   = (const float*)d_in[0];   // [N,32]
  const float* x_global  = (const float*)d_in[1];   // [N,16]
  const float* edge_f    = (const float*)d_in[2];   // [E,10]
  const int*   ei        = (const int*)  d_in[3];   // [2,E]
  const int*   src = ei;
  const int*   dst = ei + Ee;
  const float* ew_fc1_w = (const float*)d_in[4];
  const float* ew_fc1_b = (const float*)d_in[5];
  const float* ew_fc2_w = (const float*)d_in[6];
  const float* ew_fc2_b = (const float*)d_in[7];
  const float* gat0_W   = (const float*)d_in[8];
  const float* gat0_as  = (const float*)d_in[9];
  const float* gat0_ad  = (const float*)d_in[10];
  // d_in[11] = gat0_b : cancels in BatchNorm, unused
  const float* bn0_g    = (const float*)d_in[12];
  const float* bn0_b    = (const float*)d_in[13];
  const float* gat1_W   = (const float*)d_in[14];
  const float* gat1_as  = (const float*)d_in[15];
  const float* gat1_ad  = (const float*)d_in[16];
  // d_in[17] = gat1_b : cancels in BatchNorm, unused
  const float* bn1_g    = (const float*)d_in[18];
  const float* bn1_b    = (const float*)d_in[19];
  const float* la1_w    = (const float*)d_in[20];
  const float* la1_b    = (const float*)d_in[21];
  const float* la2_w    = (const float*)d_in[22];
  const float* la2_b    = (const float*)d_in[23];
  const float* gcn_W    = (const float*)d_in[24];
  const float* gcn_b    = (const float*)d_in[25];
  const float* ln_g     = (const float*)d_in[26];
  const float* ln_b     = (const float*)d_in[27];
  const float* ga1_w    = (const float*)d_in[28];
  const float* ga1_b    = (const float*)d_in[29];
  const float* ga2_w    = (const float*)d_in[30];
  const float* ga2_b    = (const float*)d_in[31];
  const float* fa1_w    = (const float*)d_in[32];
  const float* fa1_b    = (const float*)d_in[33];
  const float* fa2_w    = (const float*)d_in[34];
  const float* fa2_b    = (const float*)d_in[35];
  const float* gate_w   = (const float*)d_in[36];
  const float* gate_b   = (const float*)d_in[37];
  const float* out_w    = (const float*)d_in[38];
  const float* out_b    = (const float*)d_in[39];

  // ---- workspace carve-out (floats) ----
  float* ws = (float*)d_ws;
  size_t off = 0;
  auto carve = [&](size_t n){ float* p = ws + off; off += n; return p; };
  float* ew    = carve(Ee);
  float* exb   = carve(ESL);
  float* sA    = carve(Nn);
  float* tA    = carve(Nn);
  float* mx    = carve(Nn);
  float* dn    = carve(Nn);
  float* deg   = carve(Nn);
  float* dinvv = carve(Nn);
  float* stats = carve(256);                 // cs[64] cq[64] scale[64] shift[64]
  float* B1 = carve((size_t)Nn * 64);        // z (both GAT layers), later gate
  float* B2 = carve((size_t)Nn * 64);        // aggregation
  float* B3 = carve((size_t)Nn * 64);        // h0 -> hl
  float* B4 = carve((size_t)Nn * 64);        // zg, later fa_t
  float* B5 = carve((size_t)Nn * 64);        // hg
  float* B6 = carve((size_t)Nn * 128);       // cat
  float* B8 = carve((size_t)Nn * 32);        // adapter mid

  const int B = 256;
  auto gb = [](size_t n){ return (unsigned)((n + 255) / 256); };
  // one wave per 16x16 tile; grid = tiles * 32 threads
#define GEMM(KK, KO, MODE, ACC, A, W, bias, C)                                   \
  k_gemm<KK, KO, MODE, ACC>                                                      \
      <<<gb((size_t)(Nn / 16) * ((KO) / 16) * 32), B, 0, stream>>>(A, W, bias, C)

  float* pred  = (float*)d_out;
  float* fused = (float*)d_out + Nn;

  // 1) dynamic edge weights
  k_edgew<<<gb(Ee), B, 0, stream>>>(edge_f, ew_fc1_w, ew_fc1_b, ew_fc2_w, ew_fc2_b, ew);

  // 2) local GAT layer 0
  GEMM(32, 64, 0, 0, x_local, gat0_W, nullptr, B1);
  k_scores<<<gb(Nn), B, 0, stream>>>(B1, gat0_as, gat0_ad, sA, tA, mx);
  k_fill<<<gb(Nn), B, 0, stream>>>(dn, 0.f, Nn);
  k_attnmax<<<gb(Ee), B, 0, stream>>>(src, dst, sA, tA, mx);
  k_attnexp<<<gb(ESL), B, 0, stream>>>(src, dst, sA, tA, mx, exb, dn);
  k_fill<<<gb((size_t)Nn * 64), B, 0, stream>>>(B2, 0.f, (size_t)Nn * 64);
  k_gatscatter<<<gb((size_t)ESL * 64), B, 0, stream>>>(src, dst, exb, dn, B1, B2);
  k_fill<<<1, 128, 0, stream>>>(stats, 0.f, 128);
  k_colstats<<<1024, B, 0, stream>>>(B2, stats, stats + 64);
  k_bnfin<<<1, 64, 0, stream>>>(stats, stats + 64, bn0_g, bn0_b, stats + 128, stats + 192);
  k_bnelu<<<gb((size_t)Nn * 64), B, 0, stream>>>(B2, stats + 128, stats + 192, B3);

  // 3) local GAT layer 1
  GEMM(64, 64, 0, 0, B3, gat1_W, nullptr, B1);
  k_scores<<<gb(Nn), B, 0, stream>>>(B1, gat1_as, gat1_ad, sA, tA, mx);
  k_fill<<<gb(Nn), B, 0, stream>>>(dn, 0.f, Nn);
  k_attnmax<<<gb(Ee), B, 0, stream>>>(src, dst, sA, tA, mx);
  k_attnexp<<<gb(ESL), B, 0, stream>>>(src, dst, sA, tA, mx, exb, dn);
  k_fill<<<gb((size_t)Nn * 64), B, 0, stream>>>(B2, 0.f, (size_t)Nn * 64);
  k_gatscatter<<<gb((size_t)ESL * 64), B, 0, stream>>>(src, dst, exb, dn, B1, B2);
  k_fill<<<1, 128, 0, stream>>>(stats, 0.f, 128);
  k_colstats<<<1024, B, 0, stream>>>(B2, stats, stats + 64);
  k_bnfin<<<1, 64, 0, stream>>>(stats, stats + 64, bn1_g, bn1_b, stats + 128, stats + 192);
  k_bnelu<<<gb((size_t)Nn * 64), B, 0, stream>>>(B2, stats + 128, stats + 192, B3);

  // 4) local adapter: hl += relu(hl@la1+b)@la2 + b
  GEMM(64, 32, 1, 0, B3, la1_w, la1_b, B8);
  GEMM(32, 64, 0, 1, B8, la2_w, la2_b, B3);

  // 5) global GCN branch
  k_fill<<<gb(Nn), B, 0, stream>>>(deg, 1.f, Nn);          // self-loop weight 1
  k_deg<<<gb(Ee), B, 0, stream>>>(dst, ew, deg);
  k_dinv<<<gb(Nn), B, 0, stream>>>(deg, dinvv);
  GEMM(16, 64, 0, 0, x_global, gcn_W, nullptr, B4);
  k_fill<<<gb((size_t)Nn * 64), B, 0, stream>>>(B5, 0.f, (size_t)Nn * 64);
  k_gcnscatter<<<gb((size_t)ESL * 64), B, 0, stream>>>(src, dst, ew, dinvv, B4, B5);
  k_lnrelu<<<gb((size_t)Nn * 32), B, 0, stream>>>(B5, gcn_b, ln_g, ln_b);
  GEMM(64, 32, 1, 0, B5, ga1_w, ga1_b, B8);
  GEMM(32, 64, 0, 1, B8, ga2_w, ga2_b, B5);

  // 6) gated fusion + fusion adapter + output head
  k_cat<<<gb((size_t)Nn * 128), B, 0, stream>>>(B3, B5, B6);
  GEMM(128, 64, 2, 0, B6, gate_w, gate_b, B1);             // sigmoid gate
  GEMM(128, 64, 1, 0, B6, fa1_w, fa1_b, B4);               // relu(cat@fa1+b)
  k_fuse<<<gb((size_t)Nn * 64), B, 0, stream>>>(B1, B3, B5, fused);
  GEMM(64, 64, 0, 1, B4, fa2_w, fa2_b, fused);             // fused += t@fa2+b
  k_out<<<gb((size_t)Nn * 32), B, 0, stream>>>(fused, out_w, out_b, pred);
#undef GEMM
}